// clDiceLoss_40329742910150
// MI455X (gfx1250) — compile-verified
//
#include <hip/hip_runtime.h>
#include <hip/hip_bf16.h>

// clDice loss for (16,1,1024,1024) f32 on MI455X (gfx1250).
// HBM-roofline design (~250us): per soft-skeletonize iteration, one fused
// LDS-tiled stencil kernel using CDNA5 async global->LDS loads, with
// register-rolling 4-wide strips so LDS traffic (~28B/px) stays under the
// ~4.4x LDS:HBM bandwidth ratio. Reductions use V_WMMA_F32_16X16X4_F32.

#define IMG_N 16
#define IMG_H 1024
#define IMG_W 1024

#define TX 124           // output tile width  (keeps 16B-aligned f4 stores)
#define TY 64            // output tile height (1024/64 exact)
#define XSW 128          // xs cols = TX+4, padded stride (16B-aligned rows)
#define XSH 68           // xs rows = TY+4
#define MSW 128          // ms stride (126 used cols, padded)
#define MSH 66           // ms rows = TY+2
#define THREADS 256
#define SKEL_ITERS 20
#define NXT 9            // ceil(1024/124) x-tiles

typedef float v2f __attribute__((ext_vector_type(2)));
typedef float v8f __attribute__((ext_vector_type(8)));

// ---- CDNA5 async global->LDS copy (ASYNCcnt path) ----
__device__ __forceinline__ void async_load_f32_to_lds(const float* gptr, void* lds_ptr) {
  unsigned lds_off = (unsigned)(size_t)lds_ptr;   // low 32 bits = LDS byte offset
  unsigned long long ga = (unsigned long long)(size_t)gptr;
  asm volatile("global_load_async_to_lds_b32 %0, %1, off"
               :: "v"(lds_off), "v"(ga)
               : "memory");
}
__device__ __forceinline__ void wait_async_all() {
  asm volatile("s_wait_asynccnt 0x0" ::: "memory");
}

// One soft-skeletonize iteration:
//   m = minpool3(x) (pad +inf);  out = relu(x - (maxpool3(m) - m)) (pad -inf)
__global__ __launch_bounds__(THREADS) void skel_iter_kernel(
    const float* __restrict__ src, float* __restrict__ dst) {
  __shared__ float xs[XSH * XSW + 8];  // x tile + halo2  (34.8 KB)
  __shared__ float ms[MSH * MSW + 8];  // minpool + halo1 (33.8 KB)

  const int tid = threadIdx.x;
  const int tx0 = blockIdx.x * TX;
  const int ty0 = blockIdx.y * TY;
  const size_t img_off = (size_t)blockIdx.z * ((size_t)IMG_H * IMG_W);
  const float* base = src + img_off;
  const float NINF = -__builtin_inff();

  // ---- async-load 68x128 halo tile; clamped coords (exact for min-pool,
  // -inf mask is applied at the ms stage). lane->lx contiguous => coalesced.
  {
    const int lx = tid & 127;
    const int lyb = tid >> 7;                       // 0 or 1
    int gx = tx0 + lx - 2;
    gx = gx < 0 ? 0 : (gx > IMG_W - 1 ? IMG_W - 1 : gx);
    const float* colbase = base + gx;
#pragma unroll 4
    for (int k = 0; k < 34; ++k) {                  // 256*34 == 68*128
      int r = lyb + 2 * k;                          // xs row 0..67
      int gy = ty0 + r - 2;
      gy = gy < 0 ? 0 : (gy > IMG_H - 1 ? IMG_H - 1 : gy);
      async_load_f32_to_lds(colbase + (size_t)gy * IMG_W, &xs[r * XSW + lx]);
    }
  }
  wait_async_all();
  __syncthreads();

  const int strip = tid & 31;        // 32 strips of 4 columns
  const int grp = tid >> 5;          // 8 row-groups (one wave each)
  const int c0 = strip * 4;

  // ---- min pool: rolling row-mins, one LDS read pass, one write pass.
  // ms[p][j] = min xs[p..p+2][j..j+2], masked to -inf outside the image.
  {
    const int p0 = (MSH * grp) >> 3;               // 66*g/8
    const int p1 = (MSH * (grp + 1)) >> 3;
    const int gyb = ty0 - 1;                       // gy of ms row p is gyb+p
    const int gx0 = tx0 + c0 - 1;
    const bool cx0 = (unsigned)(gx0 + 0) < (unsigned)IMG_W;
    const bool cx1 = (unsigned)(gx0 + 1) < (unsigned)IMG_W;
    const bool cx2 = (unsigned)(gx0 + 2) < (unsigned)IMG_W;
    const bool cx3 = (unsigned)(gx0 + 3) < (unsigned)IMG_W;
    float4 ra, rb;
    for (int r = p0; r < p1 + 2; ++r) {            // xs rows p0..p1+1
      const float* row = &xs[r * XSW + c0];
      float4 v = *(const float4*)row;              // cols c0..c0+3 (16B aligned)
      float2 w = *(const float2*)(row + 4);        // cols c0+4..c0+5
      float4 rm;
      rm.x = fminf(fminf(v.x, v.y), v.z);
      rm.y = fminf(fminf(v.y, v.z), v.w);
      rm.z = fminf(fminf(v.z, v.w), w.x);
      rm.w = fminf(fminf(v.w, w.x), w.y);
      if (r >= p0 + 2) {
        int p = r - 2;
        float4 o;
        o.x = fminf(fminf(ra.x, rb.x), rm.x);
        o.y = fminf(fminf(ra.y, rb.y), rm.y);
        o.z = fminf(fminf(ra.z, rb.z), rm.z);
        o.w = fminf(fminf(ra.w, rb.w), rm.w);
        bool iny = (unsigned)(gyb + p) < (unsigned)IMG_H;
        o.x = (iny && cx0) ? o.x : NINF;
        o.y = (iny && cx1) ? o.y : NINF;
        o.z = (iny && cx2) ? o.z : NINF;
        o.w = (iny && cx3) ? o.w : NINF;
        *(float4*)&ms[p * MSW + c0] = o;
      }
      ra = rb; rb = rm;
    }
  }
  __syncthreads();

  // ---- max pool + contour + relu-sub, rolling row-maxes.
  {
    float* obase = dst + img_off;
    const int o0 = grp * 8;                        // out rows o0..o0+7
    const int gx0 = tx0 + c0;
    const bool do_store = (c0 < TX) && (gx0 < IMG_W);  // strip 31 = halo only
    float4 ma, mb, ctrb;
    for (int q = o0; q < o0 + 10; ++q) {           // ms rows o0..o0+9
      const float* row = &ms[q * MSW + c0];
      float4 v = *(const float4*)row;
      float2 w = *(const float2*)(row + 4);
      float4 mx;
      mx.x = fmaxf(fmaxf(v.x, v.y), v.z);
      mx.y = fmaxf(fmaxf(v.y, v.z), v.w);
      mx.z = fmaxf(fmaxf(v.z, v.w), w.x);
      mx.w = fmaxf(fmaxf(v.w, w.x), w.y);
      float4 ctr;                                  // ms cols c0+1..c0+4
      ctr.x = v.y; ctr.y = v.z; ctr.z = v.w; ctr.w = w.x;
      if (q >= o0 + 2) {
        int oy = q - 2;
        float4 M;
        M.x = fmaxf(fmaxf(ma.x, mb.x), mx.x);
        M.y = fmaxf(fmaxf(ma.y, mb.y), mx.y);
        M.z = fmaxf(fmaxf(ma.z, mb.z), mx.z);
        M.w = fmaxf(fmaxf(ma.w, mb.w), mx.w);
        // xv = xs[oy+2][c0+2..c0+5]; note oy+2 == q
        const float* xr = &xs[q * XSW + c0 + 2];
        float2 x01 = *(const float2*)xr;           // 8B aligned
        float2 x23 = *(const float2*)(xr + 2);
        float4 o;                                  // relu(x - (M - mc))
        o.x = x01.x - (M.x - ctrb.x); o.x = o.x > 0.f ? o.x : 0.f;
        o.y = x01.y - (M.y - ctrb.y); o.y = o.y > 0.f ? o.y : 0.f;
        o.z = x23.x - (M.z - ctrb.z); o.z = o.z > 0.f ? o.z : 0.f;
        o.w = x23.y - (M.w - ctrb.w); o.w = o.w > 0.f ? o.w : 0.f;
        if (do_store) {
          *(float4*)&obase[(size_t)(ty0 + oy) * IMG_W + gx0] = o;  // 16B aligned
        }
      }
      ma = mb; mb = mx; ctrb = ctr;
    }
  }
}

// out2[0] += sum(skel*other); out2[1] += sum(skel).
// Wave reduction via V_WMMA_F32_16X16X4_F32 with all-ones B (f32-exact tree).
__global__ __launch_bounds__(THREADS) void reduce_pair_kernel(
    const float* __restrict__ skel, const float* __restrict__ other,
    float* __restrict__ out2) {
  const size_t N = (size_t)IMG_N * IMG_H * IMG_W;
  const size_t stride = (size_t)gridDim.x * blockDim.x;
  float p = 0.f, q = 0.f;
  for (size_t i = (size_t)blockIdx.x * blockDim.x + threadIdx.x; i < N; i += stride) {
    float s = skel[i];
    p = fmaf(s, other[i], p);
    q += s;
  }

  v2f ap = {p, 0.f};
  v2f aq = {q, 0.f};
  v2f b  = {1.f, 1.f};
  v8f c  = {};
  v8f dp = __builtin_amdgcn_wmma_f32_16x16x4_f32(false, ap, false, b, (short)0, c, false, false);
  v8f dq = __builtin_amdgcn_wmma_f32_16x16x4_f32(false, aq, false, b, (short)0, c, false, false);
  float sp = dp[0] + dp[1] + dp[2] + dp[3] + dp[4] + dp[5] + dp[6] + dp[7];
  float sq = dq[0] + dq[1] + dq[2] + dq[3] + dq[4] + dq[5] + dq[6] + dq[7];
  sp += __shfl_xor(sp, 16);   // rows 0-7 (lanes 0-15) + rows 8-15 (lanes 16-31)
  sq += __shfl_xor(sq, 16);

  __shared__ float rp[THREADS / 32], rq[THREADS / 32];
  int wv = threadIdx.x >> 5;
  if ((threadIdx.x & 31) == 0) { rp[wv] = sp; rq[wv] = sq; }
  __syncthreads();
  if (threadIdx.x == 0) {
    float tp = 0.f, tq = 0.f;
    for (int j = 0; j < THREADS / 32; ++j) { tp += rp[j]; tq += rq[j]; }
    atomicAdd(&out2[0], tp);
    atomicAdd(&out2[1], tq);
  }
}

__global__ void finalize_kernel(const float* __restrict__ sums, float* __restrict__ out) {
  float iflat = (sums[0] + 1.0f) / (sums[1] + 1.0f);   // (cl_pred . gt)
  float tflat = (sums[2] + 1.0f) / (sums[3] + 1.0f);   // (tgt_skel . pred)
  float inter = iflat * tflat;
  out[0] = 1.0f - 2.0f * inter / (iflat + tflat);
}

extern "C" void kernel_launch(void* const* d_in, const int* in_sizes, int n_in,
                              void* d_out, int out_size, void* d_ws, size_t ws_size,
                              hipStream_t stream) {
  const float* pred = (const float*)d_in[0];
  const float* gt   = (const float*)d_in[1];
  float* out  = (float*)d_out;
  float* sums = (float*)d_ws;                       // 4 f32 accumulators
  const size_t ELEMS = (size_t)IMG_N * IMG_H * IMG_W;
  float* bufA = (float*)((char*)d_ws + 256);        // 64 MB ping
  float* bufB = bufA + ELEMS;                       // 64 MB pong

  hipMemsetAsync(sums, 0, 4 * sizeof(float), stream);

  dim3 sg(NXT, IMG_H / TY, IMG_N);                  // (9,16,16)
  dim3 sb(THREADS);

  // skeletonize pred, then S1 = cl_pred.gt, S2 = sum(cl_pred)
  {
    const float* src = pred;
    float* dst = bufA;
    for (int it = 0; it < SKEL_ITERS; ++it) {
      skel_iter_kernel<<<sg, sb, 0, stream>>>(src, dst);
      src = dst;
      dst = (dst == bufA) ? bufB : bufA;
    }
    reduce_pair_kernel<<<2048, THREADS, 0, stream>>>(src, gt, sums + 0);
  }
  // skeletonize gt, then S3 = tgt_skel.pred, S4 = sum(tgt_skel)
  {
    const float* src = gt;
    float* dst = bufA;
    for (int it = 0; it < SKEL_ITERS; ++it) {
      skel_iter_kernel<<<sg, sb, 0, stream>>>(src, dst);
      src = dst;
      dst = (dst == bufA) ? bufB : bufA;
    }
    reduce_pair_kernel<<<2048, THREADS, 0, stream>>>(src, pred, sums + 2);
  }

  finalize_kernel<<<1, 1, 0, stream>>>(sums, out);
}